// MultiHeadAttentionBase_65481071406520
// MI455X (gfx1250) — compile-verified
//
#include <hip/hip_runtime.h>

typedef __attribute__((ext_vector_type(2))) float v2f;
typedef __attribute__((ext_vector_type(4))) float v4f;
typedef __attribute__((ext_vector_type(8))) float v8f;

#define SEQ   2048
#define DK    64
#define PITCH 2064          // 2048 + 16 floats: kills LDS bank conflicts on column writes
#define NT    (SEQ / 16)    // 128 column tiles
#define WAVES 8

__global__ __launch_bounds__(256)
void mha_scores_softmax_kernel(const float* __restrict__ q,
                               const float* __restrict__ k,
                               const int*   __restrict__ mask,
                               float*       __restrict__ out)
{
    extern __shared__ float smem[];            // 16 x PITCH f32 score strip (132 KB)

    const int bh   = blockIdx.y;
    const int q0   = blockIdx.x * 16;          // first query row of this block
    const int tid  = threadIdx.x;
    const int wave = tid >> 5;
    const int lane = tid & 31;
    const int r    = lane & 15;                // row-within-16 / col-within-16
    const int hi   = lane >> 4;                // 0: K{0,1}, 1: K{2,3} per ISA A/B layout

    // ---- Preload Q A-fragments: 16 chunks of K=4 (float2 per lane per chunk) ----
    const v2f* qbase = (const v2f*)q + (size_t)(bh * SEQ + q0 + r) * (DK / 2) + hi;
    v2f qa[16];
#pragma unroll
    for (int kk = 0; kk < 16; ++kk) qa[kk] = qbase[2 * kk];

    // ---- QK^T over column tiles via V_WMMA_F32_16X16X4_F32 ----
    for (int j = wave; j < NT; j += WAVES) {
        const int n0 = j * 16;
        const v2f* kbase = (const v2f*)k + (size_t)(bh * SEQ + n0 + r) * (DK / 2) + hi;
        v2f kb[16];
#pragma unroll
        for (int kk = 0; kk < 16; ++kk) kb[kk] = kbase[2 * kk];

        v8f c = {};
#pragma unroll
        for (int kk = 0; kk < 16; ++kk)
            c = __builtin_amdgcn_wmma_f32_16x16x4_f32(
                    /*neg_a=*/false, qa[kk], /*neg_b=*/false, kb[kk],
                    /*c_mod=*/(short)0, c, /*reuse_a=*/false, /*reuse_b=*/false);

        // key-padding mask depends only on N = n0 + r (one column per lane)
        const int mbit = mask[bh * SEQ + n0 + r];
#pragma unroll
        for (int i = 0; i < 8; ++i) {
            const float v = (mbit == 0) ? -1e9f : c[i] * 0.125f;   // 1/sqrt(64)
            smem[(i + 8 * hi) * PITCH + n0 + r] = v;               // M = i+8*hi, N = n0+r
        }
    }
    __syncthreads();

    // ---- Row-wise softmax: each wave owns 2 of the 16 rows ----
    for (int m = wave * 2; m < wave * 2 + 2; ++m) {
        float* row = smem + m * PITCH;

        float mx = -3.402823466e38f;
        for (int cix = lane; cix < SEQ; cix += 32) mx = fmaxf(mx, row[cix]);
#pragma unroll
        for (int off = 16; off > 0; off >>= 1) mx = fmaxf(mx, __shfl_xor(mx, off, 32));

        float sum = 0.0f;
        for (int cix = lane; cix < SEQ; cix += 32) {
            const float e = __expf(row[cix] - mx);
            row[cix] = e;
            sum += e;
        }
#pragma unroll
        for (int off = 16; off > 0; off >>= 1) sum += __shfl_xor(sum, off, 32);

        const float inv = 1.0f / sum;
        float* orow = out + ((size_t)(bh * SEQ) + q0 + m) * SEQ;
        for (int cix = lane * 4; cix < SEQ; cix += 128) {       // 128-bit coalesced stores
            v4f v = *(const v4f*)(row + cix);
            v *= inv;
            *(v4f*)(orow + cix) = v;
        }
    }
}

extern "C" void kernel_launch(void* const* d_in, const int* in_sizes, int n_in,
                              void* d_out, int out_size, void* d_ws, size_t ws_size,
                              hipStream_t stream) {
    (void)in_sizes; (void)n_in; (void)out_size; (void)d_ws; (void)ws_size;
    const float* q    = (const float*)d_in[0];
    const float* k    = (const float*)d_in[1];
    // d_in[2] = v is unused by the reference (it returns softmax probabilities)
    const int*   mask = (const int*)d_in[3];
    float*       out  = (float*)d_out;

    dim3 grid(SEQ / 16, 32);     // 128 query blocks x 32 (batch*heads)
    dim3 block(256);             // 8 wave32s per workgroup
    size_t smem_bytes = (size_t)16 * PITCH * sizeof(float);   // 132 KB LDS strip
    mha_scores_softmax_kernel<<<grid, block, smem_bytes, stream>>>(q, k, mask, out);
}